// MinGRU_46737834115317
// MI455X (gfx1250) — compile-verified
//
#include <hip/hip_runtime.h>
#include <math.h>

typedef __attribute__((ext_vector_type(2))) float v2f;
typedef __attribute__((ext_vector_type(8))) float v8f;

#define B_DIM 4
#define S_DIM 4096
#define D_DIM 1024
#define M_DIM (B_DIM * S_DIM)   /* 16384 rows of x */
#define N_DIM (2 * D_DIM)       /* 2048 cols of hg */
#define K_DIM D_DIM             /* 1024 reduction */

#define NCH   8                 /* scan chunks per channel */
#define CHL   (S_DIM / NCH)     /* 512 steps per chunk */
#define NCHAN (B_DIM * D_DIM)   /* 4096 independent channels */

__device__ __forceinline__ float sigmoid_f(float v) {
    return 1.0f / (1.0f + expf(-v));
}

// -----------------------------------------------------------------------------
// GEMM: hg[m][e] = sum_k x[m][k] * W[e][k]   (both operands K-contiguous)
// One wave computes a 32(M) x 64(N) macro tile via V_WMMA_F32_16X16X4_F32:
// per k-step it loads 2 A fragments + 4 B fragments and issues 8 WMMAs,
// reusing each B fragment across both M sub-tiles (~10.7 FLOP/B from cache).
//   hidden (e <  D) -> d_out   (scanned in place later)
//   gate   (e >= D) -> d_ws
// -----------------------------------------------------------------------------
__global__ __launch_bounds__(256)
void mingru_gemm(const float* __restrict__ x, const float* __restrict__ w,
                 float* __restrict__ hidden, float* __restrict__ gate)
{
    const int lane = threadIdx.x & 31;
    const int wave = blockIdx.x * (blockDim.x >> 5) + (threadIdx.x >> 5);
    const int NT   = N_DIM / 64;          // 32 macro tiles across N
    const int mt   = wave / NT;
    const int nt   = wave - mt * NT;
    const int m0   = mt << 5;             // 32-row macro tile
    const int n0   = nt << 6;             // 64-col macro tile

    // fp32 16x4 A fragment layout: lane%16 = M row, (lane/16)*2 = K pair base.
    const int row = lane & 15;
    const int kq  = (lane >> 4) << 1;     // 0 or 2

    const float* xp0 = x + (size_t)(m0 +  0 + row) * K_DIM + kq;
    const float* xp1 = x + (size_t)(m0 + 16 + row) * K_DIM + kq;
    const float* wp0 = w + (size_t)(n0 +  0 + row) * K_DIM + kq;
    const float* wp1 = w + (size_t)(n0 + 16 + row) * K_DIM + kq;
    const float* wp2 = w + (size_t)(n0 + 32 + row) * K_DIM + kq;
    const float* wp3 = w + (size_t)(n0 + 48 + row) * K_DIM + kq;

    v8f acc00 = {}, acc01 = {}, acc02 = {}, acc03 = {};
    v8f acc10 = {}, acc11 = {}, acc12 = {}, acc13 = {};

#pragma unroll 2
    for (int k = 0; k < K_DIM; k += 4) {
        v2f a0 = *(const v2f*)(xp0 + k);
        v2f a1 = *(const v2f*)(xp1 + k);
        v2f b0 = *(const v2f*)(wp0 + k);
        v2f b1 = *(const v2f*)(wp1 + k);
        v2f b2 = *(const v2f*)(wp2 + k);
        v2f b3 = *(const v2f*)(wp3 + k);
        acc00 = __builtin_amdgcn_wmma_f32_16x16x4_f32(false, a0, false, b0, (short)0, acc00, false, false);
        acc01 = __builtin_amdgcn_wmma_f32_16x16x4_f32(false, a0, false, b1, (short)0, acc01, false, false);
        acc02 = __builtin_amdgcn_wmma_f32_16x16x4_f32(false, a0, false, b2, (short)0, acc02, false, false);
        acc03 = __builtin_amdgcn_wmma_f32_16x16x4_f32(false, a0, false, b3, (short)0, acc03, false, false);
        acc10 = __builtin_amdgcn_wmma_f32_16x16x4_f32(false, a1, false, b0, (short)0, acc10, false, false);
        acc11 = __builtin_amdgcn_wmma_f32_16x16x4_f32(false, a1, false, b1, (short)0, acc11, false, false);
        acc12 = __builtin_amdgcn_wmma_f32_16x16x4_f32(false, a1, false, b2, (short)0, acc12, false, false);
        acc13 = __builtin_amdgcn_wmma_f32_16x16x4_f32(false, a1, false, b3, (short)0, acc13, false, false);
    }

    // hidden-vs-gate split is uniform per macro tile (n0 multiple of 64):
    // pick destination scalarly so the store loops stay branch/select free.
    float* dst;
    int colbase;
    if (n0 < D_DIM) { dst = hidden; colbase = n0; }
    else            { dst = gate;   colbase = n0 - D_DIM; }

    // C/D layout: VGPR r holds (M = r + 8*(lane>=16), N = lane%16).
    const int cmhi = (lane >> 4) << 3;   // 0 or 8
    const int cn   = lane & 15;
    v8f accs[2][4] = {{acc00, acc01, acc02, acc03}, {acc10, acc11, acc12, acc13}};

#pragma unroll
    for (int i = 0; i < 2; ++i) {
        const int cm = m0 + (i << 4) + cmhi;
#pragma unroll
        for (int j = 0; j < 4; ++j) {
            const size_t col = (size_t)(colbase + (j << 4) + cn);
#pragma unroll
            for (int r = 0; r < 8; ++r) {
                dst[(size_t)(cm + r) * D_DIM + col] = accs[i][j][r];
            }
        }
    }
}

// -----------------------------------------------------------------------------
// Scan pass 1: per (chunk, channel) local recurrence with h_local = 0 at chunk
// start. Writes local h in place into `h` (d_out) and saves the per-chunk
// carry pair (A = prod of coeffs, E = local tail value).
// -----------------------------------------------------------------------------
__global__ __launch_bounds__(256)
void mingru_scan_pass1(float* __restrict__ h, const float* __restrict__ gate,
                       float* __restrict__ carryA, float* __restrict__ carryE)
{
    const int t  = blockIdx.x * blockDim.x + threadIdx.x;  // [0, NCHAN*NCH)
    const int ch = t & (NCHAN - 1);                        // channel fastest -> coalesced
    const int c  = t >> 12;                                // chunk (NCHAN = 2^12)
    const int b  = ch >> 10;
    const int d  = ch & (D_DIM - 1);

    size_t base = ((size_t)b * S_DIM + (size_t)c * CHL) * D_DIM + d;
    float hl = 0.0f;   // local recurrence value
    float P  = 1.0f;   // product of (1 - z) over this chunk

    for (int s = 0; s < CHL; ++s) {
        const size_t idx = base + (size_t)s * D_DIM;
        const float hid = h[idx];
        const float g   = gate[idx];
        const float z   = sigmoid_f(g);
        const float omz = 1.0f - z;                               // exp(-softplus(g))
        const float gv  = (hid >= 0.0f) ? (hid + 0.5f) : sigmoid_f(hid);  // g(x)
        hl = fmaf(omz, hl, z * gv);
        P *= omz;
        h[idx] = hl;
    }
    carryA[t] = P;
    carryE[t] = hl;
}

// -----------------------------------------------------------------------------
// Scan pass 2: tiny serial prefix over the NCH chunk carries of each channel.
// carryE[t] is overwritten with the *incoming* h for chunk c.
// -----------------------------------------------------------------------------
__global__ __launch_bounds__(256)
void mingru_scan_pass2(const float* __restrict__ carryA, float* __restrict__ carryE)
{
    const int ch = blockIdx.x * blockDim.x + threadIdx.x;
    if (ch >= NCHAN) return;
    float hin = 0.0f;
#pragma unroll
    for (int c = 0; c < NCH; ++c) {
        const int t   = c * NCHAN + ch;
        const float A = carryA[t];
        const float E = carryE[t];
        carryE[t] = hin;              // incoming h for this chunk
        hin = fmaf(A, hin, E);        // h at end of chunk c
    }
}

// -----------------------------------------------------------------------------
// Scan pass 3: apply the incoming carry to every element of each chunk:
//   h_true[s] = h_local[s] + hin * prod_{u<=s}(1 - z_u)
// Coefficients are recomputed from the retained gate buffer.
// -----------------------------------------------------------------------------
__global__ __launch_bounds__(256)
void mingru_scan_pass3(float* __restrict__ h, const float* __restrict__ gate,
                       const float* __restrict__ carryE)
{
    const int t  = blockIdx.x * blockDim.x + threadIdx.x;
    const int ch = t & (NCHAN - 1);
    const int c  = t >> 12;
    const float hin = carryE[t];
    if (hin == 0.0f) return;          // chunk 0 (and any exactly-zero carry)

    const int b = ch >> 10;
    const int d = ch & (D_DIM - 1);
    size_t base = ((size_t)b * S_DIM + (size_t)c * CHL) * D_DIM + d;

    float P = hin;                    // carry * running coeff product
    for (int s = 0; s < CHL; ++s) {
        const size_t idx = base + (size_t)s * D_DIM;
        const float omz = 1.0f - sigmoid_f(gate[idx]);
        P *= omz;
        h[idx] += P;
    }
}

// -----------------------------------------------------------------------------
extern "C" void kernel_launch(void* const* d_in, const int* in_sizes, int n_in,
                              void* d_out, int out_size, void* d_ws, size_t ws_size,
                              hipStream_t stream)
{
    const float* x = (const float*)d_in[0];   // [B, S, D]
    const float* w = (const float*)d_in[1];   // [2D, D]

    float* hidden = (float*)d_out;            // [M, D]; becomes final output in place
    float* gate   = (float*)d_ws;             // [M, D]  (64 MB)
    float* carryA = gate + (size_t)M_DIM * D_DIM;
    float* carryE = carryA + (size_t)NCHAN * NCH;

    // GEMM: (M/32)*(N/64) = 16384 waves, 8 waves (256 threads) per block.
    const int gemm_waves  = (M_DIM / 32) * (N_DIM / 64);
    const int gemm_blocks = gemm_waves / 8;
    mingru_gemm<<<gemm_blocks, 256, 0, stream>>>(x, w, hidden, gate);

    mingru_scan_pass1<<<(NCHAN * NCH) / 256, 256, 0, stream>>>(hidden, gate, carryA, carryE);
    mingru_scan_pass2<<<NCHAN / 256, 256, 0, stream>>>(carryA, carryE);
    mingru_scan_pass3<<<(NCHAN * NCH) / 256, 256, 0, stream>>>(hidden, gate, carryE);
}